// cep_enc_22213570855383
// MI455X (gfx1250) — compile-verified
//
#include <hip/hip_runtime.h>

// CE-encoder forward: pure HBM-bandwidth-bound stream (~1.6 GB moved, ~68us
// floor at 23.3 TB/s). No WMMA: the per-pixel 1x8x2 contraction has a
// pixel-varying B operand, so matrix pipes offer nothing over the mem bound.
// Strategy: b128 vector traffic + NT temporal hints (write-once / read-once
// data must not pollute the 192MB L2) + global_prefetch for the next batch.

typedef float v4f __attribute__((ext_vector_type(4)));

constexpr int       BN  = 4;                 // batch
constexpr int       FN  = 8;                 // frames (== CE_CODE_N, UPSAMPLE=1)
constexpr int       CN  = 3;                 // color channels
constexpr long long HW  = 720LL * 1280LL;    // 921600 spatial
constexpr long long NP  = CN * HW;           // 2764800 pixels*channels
// output tuple layout (flat, in return order)
constexpr long long OUT1 = 0;                                   // ce_blur_img_noisy (B,12,H,W)
constexpr long long OUT2 = (long long)BN * 12 * HW;             // ce_code_up_      (B,8,C,H,W,2)
constexpr long long OUT3 = OUT2 + (long long)BN * FN * NP * 2;  // ce_blur_img      (B,12,H,W)

__device__ __forceinline__ v4f ste4(v4f x) {
  v4f r;
  r.x = x.x > 0.0f ? 1.0f : 0.0f;
  r.y = x.y > 0.0f ? 1.0f : 0.0f;
  r.z = x.z > 0.0f ? 1.0f : 0.0f;
  r.w = x.w > 0.0f ? 1.0f : 0.0f;
  return r;
}

__global__ __launch_bounds__(256) void cep_enc_stream(
    const float* __restrict__ frames,   // (B, F, C, H, W)
    const float* __restrict__ cw,       // (F, C, H, W, 2)
    float* __restrict__ out)
{
  const long long p = ((long long)blockIdx.x * blockDim.x + threadIdx.x) * 4;
  if (p >= NP) return;
  const int       c  = (int)(p / HW);           // HW % 4 == 0 -> whole float4 shares c
  const long long hw = p - (long long)c * HW;

  // ---- ce_code = (ce_weight > 0): load k-interleaved pairs, de-interleave ----
  v4f ck0[FN], ck1[FN];                         // code[f][4 pixels] for k=0 / k=1
#pragma unroll
  for (int f = 0; f < FN; ++f) {
    const v4f* base = (const v4f*)(cw + ((long long)f * NP + p) * 2);
    v4f a = ste4(__builtin_nontemporal_load(base));      // p0k0 p0k1 p1k0 p1k1
    v4f b = ste4(__builtin_nontemporal_load(base + 1));  // p2k0 p2k1 p3k0 p3k1
    v4f t0 = {a.x, a.z, b.x, b.z};
    v4f t1 = {a.y, a.w, b.y, b.w};
    ck0[f] = t0;
    ck1[f] = t1;
  }

  // ---- output 2: ce_code_up_ broadcast over b (re-interleave k) ----
#pragma unroll
  for (int f = 0; f < FN; ++f) {
    v4f ia = {ck0[f].x, ck1[f].x, ck0[f].y, ck1[f].y};
    v4f ib = {ck0[f].z, ck1[f].z, ck0[f].w, ck1[f].w};
#pragma unroll
    for (int b = 0; b < BN; ++b) {
      v4f* dst = (v4f*)(out + OUT2 + (((long long)(b * FN + f)) * NP + p) * 2);
      __builtin_nontemporal_store(ia, dst);
      __builtin_nontemporal_store(ib, dst + 1);
    }
  }

  // ---- streaming reduction over f, per batch ----
#pragma unroll
  for (int b = 0; b < BN; ++b) {
    if (b + 1 < BN)  // pull next batch's frame stream into L2 while we reduce
      __builtin_prefetch(frames + ((long long)((b + 1) * FN)) * NP + p, 0, 0);

    v4f tot = {0.f, 0.f, 0.f, 0.f};
    v4f w0  = {0.f, 0.f, 0.f, 0.f};
    v4f w1  = {0.f, 0.f, 0.f, 0.f};
#pragma unroll
    for (int f = 0; f < FN; ++f) {
      v4f fr = __builtin_nontemporal_load(
          (const v4f*)(frames + ((long long)(b * FN + f)) * NP + p));
      tot += fr;
      w0  += fr * ck0[f];
      w1  += fr * ck1[f];
    }
    tot *= 0.125f;  w0 *= 0.125f;  w1 *= 0.125f;
    v4f c0 = tot - w0;
    v4f c1 = tot - w1;

    // blur-image channel layout: ch = k*6 + s*3 + c
    const long long ob = (long long)b * 12 * HW + (long long)c * HW + hw;
    __builtin_nontemporal_store(w0, (v4f*)(out + OUT1 + ob));            // k0, weighted
    __builtin_nontemporal_store(c0, (v4f*)(out + OUT1 + ob + 3 * HW));   // k0, comp
    __builtin_nontemporal_store(w1, (v4f*)(out + OUT1 + ob + 6 * HW));   // k1, weighted
    __builtin_nontemporal_store(c1, (v4f*)(out + OUT1 + ob + 9 * HW));   // k1, comp
    // clean copy (noise sigma < 1e-9 -> identical within fp32 tolerance)
    __builtin_nontemporal_store(w0, (v4f*)(out + OUT3 + ob));
    __builtin_nontemporal_store(c0, (v4f*)(out + OUT3 + ob + 3 * HW));
    __builtin_nontemporal_store(w1, (v4f*)(out + OUT3 + ob + 6 * HW));
    __builtin_nontemporal_store(c1, (v4f*)(out + OUT3 + ob + 9 * HW));
  }
}

extern "C" void kernel_launch(void* const* d_in, const int* in_sizes, int n_in,
                              void* d_out, int out_size, void* d_ws, size_t ws_size,
                              hipStream_t stream) {
  const float* frames = (const float*)d_in[0];  // (B,F,C,H,W) fp32
  const float* cw     = (const float*)d_in[1];  // (F,C,H,W,2) fp32
  float*       out    = (float*)d_out;

  const long long nthreads = NP / 4;            // 691200
  dim3 block(256);
  dim3 grid((unsigned)((nthreads + 255) / 256)); // 2700 blocks
  cep_enc_stream<<<grid, block, 0, stream>>>(frames, cw, out);
}